// CoreMLOnlineStepModule_66709432041871
// MI455X (gfx1250) — compile-verified
//
#include <hip/hip_runtime.h>
#include <math.h>

// ---------------- model constants ----------------
#define D_     512
#define FF_    2048
#define H_     8
#define KD_    64
#define HD_    64
#define LE_    12
#define LD_    4
#define S_     8
#define B_     16
#define FIN_   345
#define CACHE_ 30
#define KW_    31
#define TB_    9
#define E_     512
#define SCALING_ 0.125f   // KD^-0.5

typedef __attribute__((ext_vector_type(2))) float v2f;
typedef __attribute__((ext_vector_type(8))) float v8f;

__device__ __forceinline__ float sigm_(float x) { return 1.f / (1.f + __expf(-x)); }

// ---------------- TDM availability ----------------
#if defined(__has_builtin)
#if __has_builtin(__builtin_amdgcn_tensor_load_to_lds) && __has_builtin(__builtin_amdgcn_s_wait_tensorcnt)
#define USE_TDM 1
#endif
#endif
#ifndef USE_TDM
#define USE_TDM 0
#endif

// =================================================================
// WMMA GEMM:  out[m][n] = alpha * ACT( sum_k A[m][k]*W(k,n) + bias[n] ) + res[m][n]
// WT==0: W is K x N row-major (W[k*N+n]);  WT==1: W is N x K row-major (W[n*K+k])
// M multiple of 16. grid = (ceil(N/128), M/16), block = 256 (8 waves, wave32).
// One wave owns one 16x16 tile: V_WMMA_F32_16X16X4_F32, 4 accumulators, K-unroll 16.
// A tile is staged to LDS by the Tensor Data Mover, *double-buffered*: the TDM
// for chunk c+1 is in flight while chunk c is consumed (s_wait_tensorcnt 1 relies
// on per-wave in-order TDM completion). TDM pad feature (2 dwords per 256) gives
// a conflict-free 516-dword row stride; tensor_dim0 < tile_dim0 zero-fills the
// K tail (K=345) for free. Weights are streamed with non-temporal loads: each
// weight is used exactly once per step and the full set exceeds the 192MB L2.
// =================================================================
#define GEMM_KC 512
#define LDSTR   516
#define ABUF    (16 * LDSTR)   // floats per A buffer

// LDS layout: row r, col c lives at r*516 + c + 2*(c>=256)  (c is even in all uses)
__device__ __forceinline__ int lds_idx(int r, int c) {
  return r * LDSTR + c + ((c >> 8) << 1);
}

#if USE_TDM
typedef unsigned int u32x4 __attribute__((ext_vector_type(4)));
typedef int          i32x4 __attribute__((ext_vector_type(4)));
typedef int          i32x8 __attribute__((ext_vector_type(8)));

__device__ __forceinline__ void tdm_load_A(const float* gsrc, unsigned lds_off,
                                           int kchunk, int K)
{
  const unsigned long long ga = (unsigned long long)(size_t)gsrc;
  u32x4 g0;
  g0.x = 1u;                                                    // count=1 (valid), user mode
  g0.y = lds_off;                                               // lds_addr (bytes)
  g0.z = (unsigned)ga;                                          // global_addr[31:0]
  g0.w = (unsigned)((ga >> 32) & 0x01FFFFFFull) | (2u << 30);   // global_addr[56:32] | type=2
  i32x8 g1;
  g1[0] = (2 << 16) | (1 << 20) | (7 << 22) | (1 << 25);        // data_size=4B, pad_en, ivl=256dw, amt=2dw
  g1[1] = (kchunk & 0xFFFF) << 16;                              // tensor_dim0[15:0] (OOB -> zero fill)
  g1[2] = ((kchunk >> 16) & 0xFFFF) | (16 << 16);               // tensor_dim0[31:16] | tensor_dim1=16
  g1[3] = (GEMM_KC << 16);                                      // tensor_dim1 hi=0 | tile_dim0=512
  g1[4] = 16;                                                   // tile_dim1=16 rows, tile_dim2=0
  g1[5] = K;                                                    // tensor_dim0_stride[31:0] = row stride
  g1[6] = 0;                                                    // stride hi | dim1_stride (unused, 2D)
  g1[7] = 0;
  const i32x4 gz = {0, 0, 0, 0};
#if __clang_major__ >= 23
  const i32x8 gz8 = {0, 0, 0, 0, 0, 0, 0, 0};
  __builtin_amdgcn_tensor_load_to_lds(g0, g1, gz, gz, gz8, 0);
#else
  __builtin_amdgcn_tensor_load_to_lds(g0, g1, gz, gz, 0);
#endif
}
#endif

// one branch-free WMMA K-step (K advance of 4); weights streamed non-temporal
#define GSTEP(ACCV, KK)                                                          \
  do {                                                                           \
    const int ka_ = (KK) + kOff;                                                 \
    v2f a_, b_;                                                                  \
    a_.x = Asb[lds_idx(laneLo, ka_)];                                            \
    a_.y = Asb[lds_idx(laneLo, ka_) + 1];                                        \
    const int kg_ = kc + ka_;                                                    \
    if (WT) {                                                                    \
      b_ = __builtin_nontemporal_load((const v2f*)(W + (size_t)n * K + kg_));    \
    } else {                                                                     \
      b_.x = __builtin_nontemporal_load(W + (size_t)kg_ * N + n);                \
      b_.y = __builtin_nontemporal_load(W + (size_t)(kg_ + 1) * N + n);          \
    }                                                                            \
    ACCV = __builtin_amdgcn_wmma_f32_16x16x4_f32(false, a_, false, b_, (short)0, \
                                                 ACCV, false, false);            \
  } while (0)

template <int ACT, int WT>
__global__ __launch_bounds__(256) void gemm_wmma(
    const float* __restrict__ A, const float* __restrict__ W,
    const float* __restrict__ bias, const float* __restrict__ res,
    float* __restrict__ out, int M, int N, int K, float alpha)
{
  __shared__ float As[2 * ABUF];
  const int tid    = threadIdx.x;
  const int wave   = tid >> 5;
  const int lane   = tid & 31;
  const int laneLo = lane & 15;
  const int kOff   = (lane >> 4) << 1;          // 0 or 2 (A/B K split across lane halves)
  const int n0     = (blockIdx.x * 8 + wave) * 16;
  const int m0     = blockIdx.y * 16;
  const bool active = (n0 < N);
  const int n      = active ? (n0 + laneLo) : 0;

#if USE_TDM
  const unsigned lds_base = (unsigned)(size_t)(void*)As;
  auto tdm_issue = [&](int kcX, int bufX) {
    const int kch = (K - kcX < GEMM_KC) ? (K - kcX) : GEMM_KC;
    tdm_load_A(A + (size_t)m0 * K + kcX, lds_base + (unsigned)(bufX * (ABUF * 4)), kch, K);
  };
#endif

  v8f acc0 = {}, acc1 = {}, acc2 = {}, acc3 = {};

  int c = 0;
  for (int kc = 0; kc < K; kc += GEMM_KC, ++c) {
    const int buf = c & 1;
    const int kchunk = (K - kc < GEMM_KC) ? (K - kc) : GEMM_KC;
#if USE_TDM
    if (wave == 0) {
      if (c == 0) tdm_issue(0, 0);                      // prime the pipeline
      if (kc + GEMM_KC < K) {                           // issue next chunk ahead
        tdm_issue(kc + GEMM_KC, buf ^ 1);
        __builtin_amdgcn_s_wait_tensorcnt(1);           // in-order: chunk c done
      } else {
        __builtin_amdgcn_s_wait_tensorcnt(0);
      }
    }
#else
    for (int idx = tid; idx < 16 * GEMM_KC; idx += 256) {
      const int r = idx >> 9, cc = idx & (GEMM_KC - 1);
      As[buf * ABUF + lds_idx(r, cc)] =
          (cc < kchunk) ? A[(size_t)(m0 + r) * K + kc + cc] : 0.f;
    }
#endif
    __syncthreads();
    const float* Asb = As + buf * ABUF;
    if (active) {
      if (kc + GEMM_KC < K) {   // prefetch next weight chunk
        const float* pf = WT ? (W + (size_t)n * K + kc + GEMM_KC)
                             : (W + (size_t)(kc + GEMM_KC) * N + n);
        __builtin_prefetch(pf, 0, 1);
      }
      const int kfull = kchunk & ~3;   // all weight loads below are in-bounds
      int k = 0;
      for (; k + 16 <= kfull; k += 16) {
        GSTEP(acc0, k); GSTEP(acc1, k + 4); GSTEP(acc2, k + 8); GSTEP(acc3, k + 12);
      }
      for (; k < kfull; k += 4) { GSTEP(acc0, k); }
      if (k < kchunk) {                // select-based (branch-free) tail: only K=345
        const int ka = k + kOff;
        v2f a, b;
        a.x = Asb[lds_idx(laneLo, ka)];
        a.y = Asb[lds_idx(laneLo, ka) + 1];
        const int kg  = kc + ka;
        const int kg0 = (kg     < K) ? kg     : (K - 1);
        const int kg1 = (kg + 1 < K) ? kg + 1 : (K - 1);
        float bx, by;
        if (WT) { bx = W[(size_t)n * K + kg0]; by = W[(size_t)n * K + kg1]; }
        else    { bx = W[(size_t)kg0 * N + n]; by = W[(size_t)kg1 * N + n]; }
        b.x = (kg     < K) ? bx : 0.f;
        b.y = (kg + 1 < K) ? by : 0.f;
        acc1 = __builtin_amdgcn_wmma_f32_16x16x4_f32(false, a, false, b, (short)0,
                                                     acc1, false, false);
      }
    }
    __syncthreads();
  }

  if (active) {
    v8f acc;
#pragma unroll
    for (int i = 0; i < 8; ++i) acc[i] = (acc0[i] + acc2[i]) + (acc1[i] + acc3[i]);
    const float bv = bias ? bias[n] : 0.f;
    const int mbase = m0 + ((lane >> 4) << 3);  // C/D: vgpr i -> M=i (+8 for hi lanes)
#pragma unroll
    for (int i = 0; i < 8; ++i) {
      float v = acc[i] + bv;
      if (ACT == 1) v = fmaxf(v, 0.f);
      if (ACT == 2) v = v * sigm_(v);
      v *= alpha;
      const size_t o = (size_t)(mbase + i) * N + n;
      if (res) v += res[o];
      out[o] = v;
    }
  }
}

// ---------------- block reduction (wave32 shuffles) ----------------
__device__ __forceinline__ float block_reduce(float v, float* sh)
{
  __syncthreads();
  for (int o = 16; o; o >>= 1) v += __shfl_xor(v, o);
  const int w = threadIdx.x >> 5, l = threadIdx.x & 31;
  if (l == 0) sh[w] = v;
  __syncthreads();
  const int nw = blockDim.x >> 5;
  if (w == 0) {
    float t = (l < nw) ? sh[l] : 0.f;
    for (int o = 16; o; o >>= 1) t += __shfl_xor(t, o);
    if (l == 0) sh[0] = t;
  }
  __syncthreads();
  return sh[0];
}

__global__ void ln_rows(const float* __restrict__ x, const float* __restrict__ g,
                        const float* __restrict__ b, float* __restrict__ out, int D)
{
  __shared__ float sh[8];
  const int row = blockIdx.x;
  const float* xr = x + (size_t)row * D;
  float s = 0.f, ss = 0.f;
  for (int i = threadIdx.x; i < D; i += blockDim.x) { float v = xr[i]; s += v; ss += v * v; }
  s  = block_reduce(s, sh);
  ss = block_reduce(ss, sh);
  const float mu  = s / D;
  const float var = ss / D - mu * mu;
  const float inv = rsqrtf(var + 1e-5f);
  float* orow = out + (size_t)row * D;
  for (int i = threadIdx.x; i < D; i += blockDim.x)
    orow[i] = (xr[i] - mu) * inv * g[i] + b[i];
}

__global__ void l2norm_rows(float* __restrict__ x, int D)
{
  __shared__ float sh[8];
  float* xr = x + (size_t)blockIdx.x * D;
  float ss = 0.f;
  for (int i = threadIdx.x; i < D; i += blockDim.x) { float v = xr[i]; ss += v * v; }
  ss = block_reduce(ss, sh);
  const float inv = 1.f / fmaxf(sqrtf(ss), 1e-4f);
  for (int i = threadIdx.x; i < D; i += blockDim.x) xr[i] *= inv;
}

// ---------------- retention: state update + einsum + rms + blend ----------------
// grid = M*H, block = 64 (thread = output dim d)
__global__ void retention_step(
    const float* __restrict__ q, const float* __restrict__ k, const float* __restrict__ v,
    const float* __restrict__ old_kv, const float* __restrict__ old_scale,
    const float* __restrict__ decay, const float* __restrict__ blendp,
    float* __restrict__ new_kv, float* __restrict__ new_scale, float* __restrict__ out)
{
  __shared__ float sred[2];
  const int nh = blockIdx.x;
  const int n = nh >> 3, h = nh & 7;
  const int d = threadIdx.x;
  const float dec   = decay[h];
  const float so    = old_scale[nh];
  const float cs    = so * dec + 1.f;
  const float blend = sqrtf(so) * dec * rsqrtf(cs);
  const float invs  = rsqrtf(cs);
  const float bf    = *blendp;
  const float* qn = q + (size_t)n * D_ + h * KD_;
  const float* kn = k + (size_t)n * D_ + h * KD_;
  const float vd  = v[(size_t)n * D_ + h * HD_ + d];
  const size_t base = ((size_t)nh * HD_ + d) * KD_;
  const float* okv = old_kv + base;
  float*       nkv = new_kv + base;
  float acc = 0.f;
#pragma unroll 8
  for (int kk = 0; kk < KD_; ++kk) {
    const float o = okv[kk];
    const float c = o * blend + vd * kn[kk] * invs;
    acc += qn[kk] * c;
    nkv[kk] = o + (c - o) * bf;
  }
  float ss = acc * acc;
  for (int o = 16; o; o >>= 1) ss += __shfl_xor(ss, o);
  if ((d & 31) == 0) sred[d >> 5] = ss;
  __syncthreads();
  const float r = rsqrtf((sred[0] + sred[1]) / (float)HD_ + 1e-6f);
  out[(size_t)n * D_ + h * HD_ + d] = acc * r;
  if (d == 0) new_scale[nh] = so + (cs - so) * bf;
}

__global__ void gate_mul(const float* __restrict__ g, const float* __restrict__ y,
                         float* __restrict__ out, int n)
{
  const int i = blockIdx.x * blockDim.x + threadIdx.x;
  if (i < n) { const float gg = g[i]; out[i] = gg * sigm_(gg) * y[i]; }
}

__global__ void glu_kernel(const float* __restrict__ in, float* __restrict__ out, int n)
{
  const int i = blockIdx.x * blockDim.x + threadIdx.x;
  if (i >= n) return;
  const int m = i >> 9, d = i & 511;
  out[i] = in[(size_t)m * 1024 + d] * sigm_(in[(size_t)m * 1024 + 512 + d]);
}

// depthwise conv over 31-tap window + BN + silu + cache rotation/blend
__global__ void dwconv_kernel(
    const float* __restrict__ cache, const float* __restrict__ cur,
    const float* __restrict__ dw_w, const float* __restrict__ dw_b,
    const float* __restrict__ bn_g, const float* __restrict__ bn_b,
    const float* __restrict__ bn_m, const float* __restrict__ bn_v,
    const float* __restrict__ ingp, float* __restrict__ h_out,
    float* __restrict__ new_cv, int M)
{
  const int i = blockIdx.x * blockDim.x + threadIdx.x;
  if (i >= M * D_) return;
  const int m = i >> 9, d = i & 511;
  const float ing = *ingp;
  const float* cm = cache + (size_t)m * CACHE_ * D_ + d;
  const float* wv = dw_w + (size_t)d * KW_;
  float*      ncv = new_cv + (size_t)m * CACHE_ * D_ + d;
  const float cu  = cur[i];
  float prev = cm[0];
  float acc  = prev * wv[0];
  for (int t = 1; t < CACHE_; ++t) {
    const float c = cm[(size_t)t * D_];
    acc += c * wv[t];
    ncv[(size_t)(t - 1) * D_] = prev + (c - prev) * ing;
    prev = c;
  }
  ncv[(size_t)(CACHE_ - 1) * D_] = prev + (cu - prev) * ing;
  acc += cu * wv[KW_ - 1] + dw_b[d];
  float hn = (acc - bn_m[d]) * rsqrtf(bn_v[d] + 1e-5f) * bn_g[d] + bn_b[d];
  h_out[i] = hn * sigm_(hn);
}

__global__ void topbuf_kernel(const float* __restrict__ old_tb, const float* __restrict__ x,
                              const float* __restrict__ ingp, float* __restrict__ new_tb)
{
  const int i = blockIdx.x * blockDim.x + threadIdx.x;
  if (i >= B_ * TB_ * D_) return;
  const int d = i % D_;
  const int t = (i / D_) % TB_;
  const int m = i / (D_ * TB_);
  new_tb[i] = (t < TB_ - 1) ? old_tb[((size_t)m * TB_ + t + 1) * D_ + d]
                            : x[(size_t)m * D_ + d] * (*ingp);
}

__global__ void cnn_pack(const float* __restrict__ tb, float* __restrict__ Ap)
{
  const int i = blockIdx.x * blockDim.x + threadIdx.x;
  if (i >= B_ * TB_ * D_) return;
  const int d = i % D_;
  const int t = (i / D_) % TB_;
  const int m = i / (D_ * TB_);
  Ap[(size_t)m * (TB_ * D_) + (size_t)d * TB_ + t] = tb[i];
}

__global__ void pos_kernel(float* __restrict__ pe)
{
  const int i = blockIdx.x * blockDim.x + threadIdx.x;
  if (i >= S_ * D_) return;
  const int s = i >> 9, d = i & 511;
  const int j = d >> 1;
  const float ang = (float)s * powf(10000.f, -(float)(2 * j) / (float)D_);
  pe[i] = (d & 1) ? cosf(ang) : sinf(ang);
}

__global__ void cat_kernel(const float* __restrict__ emb, const float* __restrict__ pe,
                           float* __restrict__ A)
{
  const int i = blockIdx.x * blockDim.x + threadIdx.x;
  if (i >= B_ * S_ * (E_ + D_)) return;
  const int c = i & 1023;
  const int row = i >> 10;
  const int b = row >> 3, s = row & 7;
  A[i] = (c < E_) ? emb[(size_t)b * E_ + c] : pe[(size_t)s * D_ + (c - E_)];
}

// tiny 8x8 attention per (b,h); grid = B*H, block = 64
__global__ void mha_attn(const float* __restrict__ q, const float* __restrict__ k,
                         const float* __restrict__ v, float* __restrict__ o)
{
  __shared__ float qs[S_][HD_], ks[S_][HD_], vs[S_][HD_], att[S_][S_];
  const int bh = blockIdx.x;
  const int b = bh >> 3, h = bh & 7;
  const int t = threadIdx.x;
  for (int idx = t; idx < S_ * HD_; idx += 64) {
    const int s = idx >> 6, d = idx & 63;
    const size_t off = ((size_t)(b * S_ + s)) * D_ + h * HD_ + d;
    qs[s][d] = q[off]; ks[s][d] = k[off]; vs[s][d] = v[off];
  }
  __syncthreads();
  {
    const int i = t >> 3, j = t & 7;
    float sc = 0.f;
    for (int d = 0; d < HD_; ++d) sc += qs[i][d] * ks[j][d];
    att[i][j] = sc * 0.125f;   // 1/sqrt(64)
  }
  __syncthreads();
  if (t < S_) {
    float mx = -1e30f;
    for (int j = 0; j < S_; ++j) mx = fmaxf(mx, att[t][j]);
    float sum = 0.f;
    for (int j = 0; j < S_; ++j) { const float e = __expf(att[t][j] - mx); att[t][j] = e; sum += e; }
    const float inv = 1.f / sum;
    for (int j = 0; j < S_; ++j) att[t][j] *= inv;
  }
  __syncthreads();
  const int d = t;
  for (int i = 0; i < S_; ++i) {
    float acc = 0.f;
    for (int j = 0; j < S_; ++j) acc += att[i][j] * vs[j][d];
    o[((size_t)(b * S_ + i)) * D_ + h * HD_ + d] = acc;
  }
}

__global__ void logits_kernel(const float* __restrict__ emb, const float* __restrict__ attr,
                              const float* __restrict__ dcdp, float* __restrict__ out)
{
  __shared__ float sh[8];
  const int bs = blockIdx.x;
  const int b = bs >> 3;
  float s = 0.f;
  for (int i = threadIdx.x; i < D_; i += blockDim.x)
    s += emb[(size_t)b * D_ + i] * attr[(size_t)bs * D_ + i];
  s = block_reduce(s, sh);
  if (threadIdx.x == 0) out[bs] = s * (*dcdp);
}

// =================================================================
extern "C" void kernel_launch(void* const* d_in, const int* in_sizes, int n_in,
                              void* d_out, int out_size, void* d_ws, size_t ws_size,
                              hipStream_t stream)
{
  (void)in_sizes; (void)out_size; (void)ws_size;
  if (n_in < 87) return;

  const float* frame     = (const float*)d_in[0];
  const float* enc_kv_in = (const float*)d_in[1];
  const float* enc_sc_in = (const float*)d_in[2];
  const float* enc_cv_in = (const float*)d_in[3];
  const float* dec_kv_in = (const float*)d_in[4];
  const float* dec_sc_in = (const float*)d_in[5];
  const float* tb_in     = (const float*)d_in[6];
  const float* ingest    = (const float*)d_in[7];
  const float* decode    = (const float*)d_in[8];

  // params flattened as a JAX pytree: sorted keys, dec -> enc -> top
  int pi = 9;
  const float* d_ak_b = (const float*)d_in[pi++]; const float* d_ak_w = (const float*)d_in[pi++];
  const float* d_ao_b = (const float*)d_in[pi++]; const float* d_ao_w = (const float*)d_in[pi++];
  const float* d_aq_b = (const float*)d_in[pi++]; const float* d_aq_w = (const float*)d_in[pi++];
  const float* d_av_b = (const float*)d_in[pi++]; const float* d_av_w = (const float*)d_in[pi++];
  const float* d_ff_b1 = (const float*)d_in[pi++]; const float* d_ff_b2 = (const float*)d_in[pi++];
  const float* d_ff_w1 = (const float*)d_in[pi++]; const float* d_ff_w2 = (const float*)d_in[pi++];
  const float* d_n11_b = (const float*)d_in[pi++]; const float* d_n11_g = (const float*)d_in[pi++];
  const float* d_n21_b = (const float*)d_in[pi++]; const float* d_n21_g = (const float*)d_in[pi++];
  const float* d_n22_b = (const float*)d_in[pi++]; const float* d_n22_g = (const float*)d_in[pi++];
  const float* d_rg_b = (const float*)d_in[pi++]; const float* d_rg_w = (const float*)d_in[pi++];
  const float* d_rk_b = (const float*)d_in[pi++]; const float* d_rk_w = (const float*)d_in[pi++];
  const float* d_ro_b = (const float*)d_in[pi++]; const float* d_ro_w = (const float*)d_in[pi++];
  const float* d_rq_b = (const float*)d_in[pi++]; const float* d_rq_w = (const float*)d_in[pi++];
  const float* d_rv_b = (const float*)d_in[pi++]; const float* d_rv_w = (const float*)d_in[pi++];

  const float* e_attn_ln_b = (const float*)d_in[pi++]; const float* e_attn_ln_g = (const float*)d_in[pi++];
  const float* e_bn_b = (const float*)d_in[pi++]; const float* e_bn_g = (const float*)d_in[pi++];
  const float* e_bn_m = (const float*)d_in[pi++]; const float* e_bn_v = (const float*)d_in[pi++];
  const float* e_conv_ln_b = (const float*)d_in[pi++]; const float* e_conv_ln_g = (const float*)d_in[pi++];
  const float* e_dw_b = (const float*)d_in[pi++]; const float* e_dw_w = (const float*)d_in[pi++];
  const float* e_ff1_b1 = (const float*)d_in[pi++]; const float* e_ff1_b2 = (const float*)d_in[pi++];
  const float* e_ff1_ln_b = (const float*)d_in[pi++]; const float* e_ff1_ln_g = (const float*)d_in[pi++];
  const float* e_ff1_w1 = (const float*)d_in[pi++]; const float* e_ff1_w2 = (const float*)d_in[pi++];
  const float* e_ff2_b1 = (const float*)d_in[pi++]; const float* e_ff2_b2 = (const float*)d_in[pi++];
  const float* e_ff2_ln_b = (const float*)d_in[pi++]; const float* e_ff2_ln_g = (const float*)d_in[pi++];
  const float* e_ff2_w1 = (const float*)d_in[pi++]; const float* e_ff2_w2 = (const float*)d_in[pi++];
  const float* e_fin_b = (const float*)d_in[pi++]; const float* e_fin_g = (const float*)d_in[pi++];
  const float* e_g_b = (const float*)d_in[pi++]; const float* e_g_w = (const float*)d_in[pi++];
  const float* e_k_b = (const float*)d_in[pi++]; const float* e_k_w = (const float*)d_in[pi++];
  const float* e_o_b = (const float*)d_in[pi++]; const float* e_o_w = (const float*)d_in[pi++];
  const float* e_pw1_b = (const float*)d_in[pi++]; const float* e_pw1_w = (const float*)d_in[pi++];
  const float* e_pw2_b = (const float*)d_in[pi++]; const float* e_pw2_w = (const float*)d_in[pi++];
  const float* e_q_b = (const float*)d_in[pi++]; const float* e_q_w = (const float*)d_in[pi++];
  const float* e_v_b = (const float*)d_in[pi++]; const float* e_v_w = (const float*)d_in[pi++];

  const float* t_cnn_b = (const float*)d_in[pi++]; const float* t_cnn_w = (const float*)d_in[pi++];
  const float* t_cv_b = (const float*)d_in[pi++]; const float* t_cv_w = (const float*)d_in[pi++];
  const float* t_dec_decay = (const float*)d_in[pi++];
  const float* t_dnorm_b = (const float*)d_in[pi++]; const float* t_dnorm_g = (const float*)d_in[pi++];
  const float* t_enc_decay = (const float*)d_in[pi++];
  const float* t_in_b = (const float*)d_in[pi++];
  const float* t_in_ln_b = (const float*)d_in[pi++]; const float* t_in_ln_g = (const float*)d_in[pi++];
  const float* t_in_w = (const float*)d_in[pi++];

  // ---------------- output regions (flat return order) ----------------
  float* out = (float*)d_out;
  float* o_logits = out;                         // 128
  float* o_kv  = o_logits + 128;                 // 12*16*8*64*64
  float* o_sc  = o_kv + (size_t)LE_ * B_ * H_ * HD_ * KD_;
  float* o_cv  = o_sc + (size_t)LE_ * B_ * H_;
  float* o_dkv = o_cv + (size_t)LE_ * B_ * CACHE_ * D_;
  float* o_dsc = o_dkv + (size_t)LD_ * (B_ * S_) * H_ * HD_ * KD_;
  float* o_tb  = o_dsc + (size_t)LD_ * (B_ * S_) * H_;

  // ---------------- workspace ----------------
  float* ws = (float*)d_ws;
  size_t wo = 0;
  auto NXT = [&](size_t cnt) { float* p = ws + wo; wo += cnt; return p; };
  float* x    = NXT(16 * D_);
  float* t0   = NXT(128 * D_);          // LN scratch (16 or 128 rows)
  float* t1   = NXT(128 * FF_);         // FFN hidden / pw1 out
  float* qb   = NXT(128 * D_);
  float* kb   = NXT(128 * D_);
  float* vb   = NXT(128 * D_);
  float* gb   = NXT(128 * D_);
  float* rout = NXT(128 * D_);
  float* gmul = NXT(128 * D_);
  float* glu  = NXT(16 * D_);
  float* cvh  = NXT(16 * D_);
  float* emb  = NXT(16 * E_);
  float* Ap   = NXT(16 * TB_ * D_);
  float* pe   = NXT(S_ * D_);
  float* acat = NXT(128 * (E_ + D_));
  float* attr = NXT(128 * D_);
  float* mo   = NXT(128 * D_);

#define GEMM(ACT, WT, Ab, Wb, Bb, Rb, Ob, M, N, K, al) \
  gemm_wmma<ACT, WT><<<dim3(((N) + 127) / 128, (M) / 16), 256, 0, stream>>>(Ab, Wb, Bb, Rb, Ob, M, N, K, al)
#define EW(n) dim3(((n) + 255) / 256), dim3(256), 0, stream

  // ---------------- input projection ----------------
  GEMM(0, 0, frame, t_in_w, t_in_b, nullptr, t0, 16, D_, FIN_, 1.f);
  ln_rows<<<16, 256, 0, stream>>>(t0, t_in_ln_g, t_in_ln_b, x, D_);

  // ---------------- encoder ----------------
  for (int i = 0; i < LE_; ++i) {
    const size_t oD = (size_t)i * D_, oFF = (size_t)i * FF_;
    const size_t oDD = (size_t)i * D_ * D_, oDF = (size_t)i * D_ * FF_;
    // FFN1 (half residual)
    ln_rows<<<16, 256, 0, stream>>>(x, e_ff1_ln_g + oD, e_ff1_ln_b + oD, t0, D_);
    GEMM(2, 0, t0, e_ff1_w1 + oDF, e_ff1_b1 + oFF, nullptr, t1, 16, FF_, D_, 1.f);
    GEMM(0, 0, t1, e_ff1_w2 + oDF, e_ff1_b2 + oD, x, x, 16, D_, FF_, 0.5f);
    // retention
    ln_rows<<<16, 256, 0, stream>>>(x, e_attn_ln_g + oD, e_attn_ln_b + oD, t0, D_);
    GEMM(0, 0, t0, e_q_w + oDD, e_q_b + oD, nullptr, qb, 16, D_, D_, 1.f);
    GEMM(0, 0, t0, e_k_w + oDD, e_k_b + oD, nullptr, kb, 16, D_, D_, SCALING_);
    GEMM(0, 0, t0, e_v_w + oDD, e_v_b + oD, nullptr, vb, 16, D_, D_, 1.f);
    GEMM(0, 0, t0, e_g_w + oDD, e_g_b + oD, nullptr, gb, 16, D_, D_, 1.f);
    retention_step<<<B_ * H_, 64, 0, stream>>>(
        qb, kb, vb, enc_kv_in + (size_t)i * B_ * H_ * HD_ * KD_,
        enc_sc_in + (size_t)i * B_ * H_, t_enc_decay, ingest,
        o_kv + (size_t)i * B_ * H_ * HD_ * KD_, o_sc + (size_t)i * B_ * H_, rout);
    gate_mul<<<EW(16 * D_)>>>(gb, rout, gmul, 16 * D_);
    GEMM(0, 0, gmul, e_o_w + oDD, e_o_b + oD, x, x, 16, D_, D_, 1.f);
    // conv block
    ln_rows<<<16, 256, 0, stream>>>(x, e_conv_ln_g + oD, e_conv_ln_b + oD, t0, D_);
    GEMM(0, 1, t0, e_pw1_w + (size_t)i * 2 * D_ * D_, e_pw1_b + (size_t)i * 2 * D_,
         nullptr, t1, 16, 2 * D_, D_, 1.f);
    glu_kernel<<<EW(16 * D_)>>>(t1, glu, 16 * D_);
    dwconv_kernel<<<EW(16 * D_)>>>(
        enc_cv_in + (size_t)i * B_ * CACHE_ * D_, glu,
        e_dw_w + (size_t)i * D_ * KW_, e_dw_b + oD,
        e_bn_g + oD, e_bn_b + oD, e_bn_m + oD, e_bn_v + oD, ingest,
        cvh, o_cv + (size_t)i * B_ * CACHE_ * D_, 16);
    GEMM(0, 1, cvh, e_pw2_w + oDD, e_pw2_b + oD, x, x, 16, D_, D_, 1.f);
    // FFN2 (half residual)
    ln_rows<<<16, 256, 0, stream>>>(x, e_ff2_ln_g + oD, e_ff2_ln_b + oD, t0, D_);
    GEMM(2, 0, t0, e_ff2_w1 + oDF, e_ff2_b1 + oFF, nullptr, t1, 16, FF_, D_, 1.f);
    GEMM(0, 0, t1, e_ff2_w2 + oDF, e_ff2_b2 + oD, x, x, 16, D_, FF_, 0.5f);
    // final LN (in place)
    ln_rows<<<16, 256, 0, stream>>>(x, e_fin_g + oD, e_fin_b + oD, x, D_);
  }

  // ---------------- top: buffer shift + CNN pooling + attr build ----------------
  topbuf_kernel<<<EW(B_ * TB_ * D_)>>>(tb_in, x, ingest, o_tb);
  cnn_pack<<<EW(B_ * TB_ * D_)>>>(o_tb, Ap);
  GEMM(0, 1, Ap, t_cnn_w, t_cnn_b, nullptr, emb, 16, E_, TB_ * D_, 1.f);
  l2norm_rows<<<16, 256, 0, stream>>>(emb, E_);
  pos_kernel<<<EW(S_ * D_)>>>(pe);
  cat_kernel<<<EW(B_ * S_ * (E_ + D_))>>>(emb, pe, acat);
  GEMM(0, 0, acat, t_cv_w, t_cv_b, nullptr, attr, 128, D_, E_ + D_, 1.f);

  // ---------------- decoder fusion layers ----------------
  for (int i = 0; i < LD_; ++i) {
    const size_t oD = (size_t)i * D_, oFF = (size_t)i * FF_;
    const size_t oDD = (size_t)i * D_ * D_, oDF = (size_t)i * D_ * FF_;
    // retention (128 tokens, T=1)
    ln_rows<<<128, 256, 0, stream>>>(attr, d_n11_g + oD, d_n11_b + oD, t0, D_);
    GEMM(0, 0, t0, d_rq_w + oDD, d_rq_b + oD, nullptr, qb, 128, D_, D_, 1.f);
    GEMM(0, 0, t0, d_rk_w + oDD, d_rk_b + oD, nullptr, kb, 128, D_, D_, SCALING_);
    GEMM(0, 0, t0, d_rv_w + oDD, d_rv_b + oD, nullptr, vb, 128, D_, D_, 1.f);
    GEMM(0, 0, t0, d_rg_w + oDD, d_rg_b + oD, nullptr, gb, 128, D_, D_, 1.f);
    retention_step<<<128 * H_, 64, 0, stream>>>(
        qb, kb, vb, dec_kv_in + (size_t)i * 128 * H_ * HD_ * KD_,
        dec_sc_in + (size_t)i * 128 * H_, t_dec_decay, decode,
        o_dkv + (size_t)i * 128 * H_ * HD_ * KD_, o_dsc + (size_t)i * 128 * H_, rout);
    gate_mul<<<EW(128 * D_)>>>(gb, rout, gmul, 128 * D_);
    GEMM(0, 0, gmul, d_ro_w + oDD, d_ro_b + oD, attr, attr, 128, D_, D_, 1.f);
    // MHA over S=8
    ln_rows<<<128, 256, 0, stream>>>(attr, d_n21_g + oD, d_n21_b + oD, t0, D_);
    GEMM(0, 0, t0, d_aq_w + oDD, d_aq_b + oD, nullptr, qb, 128, D_, D_, 1.f);
    GEMM(0, 0, t0, d_ak_w + oDD, d_ak_b + oD, nullptr, kb, 128, D_, D_, 1.f);
    GEMM(0, 0, t0, d_av_w + oDD, d_av_b + oD, nullptr, vb, 128, D_, D_, 1.f);
    mha_attn<<<B_ * H_, 64, 0, stream>>>(qb, kb, vb, mo);
    GEMM(0, 0, mo, d_ao_w + oDD, d_ao_b + oD, attr, attr, 128, D_, D_, 1.f);
    // FFN (relu)
    ln_rows<<<128, 256, 0, stream>>>(attr, d_n22_g + oD, d_n22_b + oD, t0, D_);
    GEMM(1, 0, t0, d_ff_w1 + oDF, d_ff_b1 + oFF, nullptr, t1, 128, FF_, D_, 1.f);
    GEMM(0, 0, t1, d_ff_w2 + oDF, d_ff_b2 + oD, attr, attr, 128, D_, FF_, 1.f);
  }

  // ---------------- logits ----------------
  ln_rows<<<128, 256, 0, stream>>>(attr, t_dnorm_g, t_dnorm_b, t0, D_);
  l2norm_rows<<<128, 256, 0, stream>>>(t0, D_);
  logits_kernel<<<128, 256, 0, stream>>>(emb, t0, decode, o_logits);

#undef GEMM
#undef EW
}